// SparseAutoencoder_47047071760615
// MI455X (gfx1250) — compile-verified
//
#include <hip/hip_runtime.h>
#include <hip/hip_bf16.h>

typedef __attribute__((ext_vector_type(2))) float v2f;
typedef __attribute__((ext_vector_type(8))) float v8f;
typedef int sae_v4i __attribute__((vector_size(16)));   // matches builtin param type

#define BM 128
#define BN 128
#define BK 32
#define LPAD 4   // row stride 36 floats = 144 B: 16B-aligned rows (async b128 dst) + conflict-free b64 reads
#define LSTRIDE (BK + LPAD)

#if __has_builtin(__builtin_amdgcn_global_load_async_to_lds_b128)
#define SAE_HAS_ASYNC 1
#else
#define SAE_HAS_ASYNC 0
#endif

#define GLOBAL_AS __attribute__((address_space(1)))
#define LDS_AS    __attribute__((address_space(3)))

static __device__ __forceinline__ void sae_copy16(const float* g, float* l) {
#if SAE_HAS_ASYNC
    // gfx1250 async copy: global -> LDS, tracked on ASYNCcnt
    __builtin_amdgcn_global_load_async_to_lds_b128(
        (GLOBAL_AS sae_v4i*)(float*)g, (LDS_AS sae_v4i*)l, 0, 0);
#else
    *reinterpret_cast<float4*>(l) = *reinterpret_cast<const float4*>(g);
#endif
}

static __device__ __forceinline__ void sae_wait_async() {
#if SAE_HAS_ASYNC
#if __has_builtin(__builtin_amdgcn_s_wait_asynccnt)
    __builtin_amdgcn_s_wait_asynccnt(0);
#else
    asm volatile("s_wait_asynccnt 0" ::: "memory");
#endif
#endif
}

// C[M,N] = A[M,K] * Bw[N,K]^T + bias[N]
// 128 threads = 4 waves; each wave computes a 64x64 sub-tile as 4x4 WMMA tiles.
// Double-buffered LDS staging via async global->LDS copies.
__global__ __launch_bounds__(128) void sae_gemm_xt(
    const float* __restrict__ A, const float* __restrict__ Bw,
    const float* __restrict__ bias, float* __restrict__ C,
    int M, int N, int Kd)
{
    __shared__ float As[2][BM][LSTRIDE];
    __shared__ float Bs[2][BN][LSTRIDE];

    const int t    = threadIdx.x;       // 0..127
    const int wave = t >> 5;            // 0..3
    const int lane = t & 31;
    const int half = lane >> 4;         // 0 or 1
    const int lm   = lane & 15;         // 0..15
    const int wm   = wave >> 1;         // 0..1  (64-row offset)
    const int wn   = wave & 1;          // 0..1  (64-col offset)
    const int m0   = blockIdx.x * BM;
    const int n0   = blockIdx.y * BN;

    v8f acc[4][4];
    const v8f vzero = {0.f,0.f,0.f,0.f,0.f,0.f,0.f,0.f};
#pragma unroll
    for (int i = 0; i < 4; ++i)
#pragma unroll
        for (int j = 0; j < 4; ++j)
            acc[i][j] = vzero;

    // stage tile (BMxBK of A, BNxBK of Bw) for K-chunk kt into buffer buf
    auto stage = [&](int kt, int buf) {
#pragma unroll
        for (int i = 0; i < 8; ++i) {
            const int idx = t + i * 128;      // 0..1023 float4 slots
            const int row = idx >> 3;         // 0..127
            const int c4  = idx & 7;          // float4 column
            sae_copy16(A  + (size_t)(m0 + row) * Kd + kt + c4 * 4,
                       &As[buf][row][c4 * 4]);
            sae_copy16(Bw + (size_t)(n0 + row) * Kd + kt + c4 * 4,
                       &Bs[buf][row][c4 * 4]);
        }
    };

    const int nk = Kd / BK;
    stage(0, 0);
    sae_wait_async();
    __syncthreads();

    for (int s = 0; s < nk; ++s) {
        const int buf = s & 1;
        if (s + 1 < nk) stage((s + 1) * BK, buf ^ 1);  // overlap next-stage copy

        // ---- compute: 8 K-steps of 4, 16 WMMAs per step per wave ----
#pragma unroll
        for (int k4 = 0; k4 < BK; k4 += 4) {
            const int kc = k4 + 2 * half;   // lane's K pair within tile
            v2f af[4], bf[4];
#pragma unroll
            for (int i = 0; i < 4; ++i) {
                const int ar = wm * 64 + i * 16 + lm;
                af[i].x = As[buf][ar][kc];
                af[i].y = As[buf][ar][kc + 1];
                const int br = wn * 64 + i * 16 + lm;
                bf[i].x = Bs[buf][br][kc];
                bf[i].y = Bs[buf][br][kc + 1];
            }
#pragma unroll
            for (int i = 0; i < 4; ++i)
#pragma unroll
                for (int j = 0; j < 4; ++j)
                    acc[i][j] = __builtin_amdgcn_wmma_f32_16x16x4_f32(
                        false, af[i], false, bf[j], (short)0, acc[i][j],
                        false, false);
        }
        sae_wait_async();
        __syncthreads();
    }

    // ---- epilogue: D layout: lane (L%16)=N, VGPR r: M = r + 8*(L/16) ----
#pragma unroll
    for (int j = 0; j < 4; ++j) {
        const int n = n0 + wn * 64 + j * 16 + lm;
        const float bvv = bias ? bias[n] : 0.0f;
#pragma unroll
        for (int i = 0; i < 4; ++i) {
            const int mbase = m0 + wm * 64 + i * 16 + half * 8;
#pragma unroll
            for (int r = 0; r < 8; ++r) {
                C[(size_t)(mbase + r) * N + n] = acc[i][j][r] + bvv;
            }
        }
    }
}

// b_adj[j] = b_enc[j] - dot(W_enc[j,:], b_pre)   (exact bias fold of centering)
// one wave per j, 8 waves per block
__global__ __launch_bounds__(256) void sae_bias_fold(
    const float* __restrict__ W_enc, const float* __restrict__ b_enc,
    const float* __restrict__ b_pre, float* __restrict__ b_adj, int D)
{
    const int wave = threadIdx.x >> 5;
    const int lane = threadIdx.x & 31;
    const int j = blockIdx.x * 8 + wave;
    const float* wrow = W_enc + (size_t)j * D;
    float s = 0.0f;
    for (int d = lane; d < D; d += 32) s += wrow[d] * b_pre[d];
#pragma unroll
    for (int off = 16; off > 0; off >>= 1) s += __shfl_down(s, off, 32);
    if (lane == 0) b_adj[j] = b_enc[j] - s;
}

// Per-row exact top-K via 4-pass byte radix-select on monotone float keys,
// then in-place mask:  h = (key >= kth_key) ? relu(h) : 0
#define TOPK_PER 15
__global__ __launch_bounds__(256) void sae_topk_mask(
    float* __restrict__ h, int H, int Ksel)
{
    const int row = blockIdx.x;
    const int t   = threadIdx.x;
    float* hrow = h + (size_t)row * H;

    __shared__ unsigned int hist[256];
    __shared__ unsigned int s_prefix;
    __shared__ unsigned int s_want;

    unsigned int myk[TOPK_PER];
    float        myv[TOPK_PER];
#pragma unroll
    for (int i = 0; i < TOPK_PER; ++i) {
        const float v = hrow[t + i * 256];
        myv[i] = v;
        unsigned int u = __float_as_uint(v);
        u = (u & 0x80000000u) ? ~u : (u | 0x80000000u);  // monotone key
        myk[i] = u;
    }
    if (t == 0) { s_prefix = 0u; s_want = (unsigned int)Ksel; }

    for (int shift = 24; shift >= 0; shift -= 8) {
        hist[t] = 0u;
        __syncthreads();
        const unsigned int pfx = s_prefix;
        const unsigned int maskhi =
            (shift == 24) ? 0u : (0xFFFFFFFFu << (shift + 8));
#pragma unroll
        for (int i = 0; i < TOPK_PER; ++i) {
            const unsigned int u = myk[i];
            if ((u & maskhi) == pfx)
                atomicAdd(&hist[(u >> shift) & 255u], 1u);
        }
        __syncthreads();
        if (t == 0) {
            unsigned int want = s_want;
            unsigned int cum = 0u;
            for (int b = 255; b >= 0; --b) {
                const unsigned int c = hist[b];
                if (cum + c >= want) {
                    s_want   = want - cum;
                    s_prefix = pfx | ((unsigned int)b << shift);
                    break;
                }
                cum += c;
            }
        }
        __syncthreads();
    }

    const unsigned int kth = s_prefix;   // exact key of K-th largest
#pragma unroll
    for (int i = 0; i < TOPK_PER; ++i) {
        const bool keep = (myk[i] >= kth);
        hrow[t + i * 256] = keep ? fmaxf(myv[i], 0.0f) : 0.0f;
    }
}

extern "C" void kernel_launch(void* const* d_in, const int* in_sizes, int n_in,
                              void* d_out, int out_size, void* d_ws, size_t ws_size,
                              hipStream_t stream) {
    (void)in_sizes; (void)n_in; (void)out_size; (void)ws_size;
    constexpr int B = 32768, D = 768, H = 3840, K = 64;

    const float* x     = (const float*)d_in[0];
    const float* W_enc = (const float*)d_in[1];
    const float* b_enc = (const float*)d_in[2];
    const float* W_dec = (const float*)d_in[3];
    const float* b_pre = (const float*)d_in[4];

    float* out    = (float*)d_out;
    float* x_rec  = out;                      // [B, D]
    float* h_sp   = out + (size_t)B * D;      // [B, H] dense sparse-hidden
    float* b_adj  = (float*)d_ws;             // [H] folded encoder bias

    // 0) fold centering into encoder bias: b_adj = b_enc - W_enc @ b_pre
    sae_bias_fold<<<H / 8, 256, 0, stream>>>(W_enc, b_enc, b_pre, b_adj, D);

    // 1) encoder: h = x @ W_enc^T + b_adj  -> written into h_sp region
    sae_gemm_xt<<<dim3(B / BM, H / BN), 128, 0, stream>>>(
        x, W_enc, b_adj, h_sp, B, H, D);

    // 2) per-row top-K + relu, masked in place -> h_sparse
    sae_topk_mask<<<B, 256, 0, stream>>>(h_sp, H, K);

    // 3) decoder: x_rec = h_sparse @ W_dec^T + b_pre
    sae_gemm_xt<<<dim3(B / BM, D / BN), 128, 0, stream>>>(
        h_sp, W_dec, b_pre, x_rec, B, D, H);
}